// SpatialAttention_4698694221989
// MI455X (gfx1250) — compile-verified
//
#include <hip/hip_runtime.h>
#include <hip/hip_bf16.h>
#include <math.h>

// Problem dims (compile-time, from the reference)
#define BB 64
#define TT 12
#define NN 512
#define FF 64
#define UU 64

typedef __attribute__((ext_vector_type(2))) float v2f;
typedef __attribute__((ext_vector_type(8))) float v8f;
typedef __attribute__((ext_vector_type(4))) unsigned int v4u;
typedef __attribute__((ext_vector_type(8))) int v8i;
typedef __attribute__((ext_vector_type(4))) int v4i;

// ---------------------------------------------------------------------------
// TDM bulk copy: contiguous `nelem` f32 from global -> LDS via the Tensor
// Data Mover (ISA ch. 8 descriptor layout, 1-D tile).  Issue from ONE wave,
// then s_wait_tensorcnt 0 + workgroup barrier before consumers read LDS.
// ---------------------------------------------------------------------------
__device__ __forceinline__ void tdm_load_f32_to_lds(const float* gptr,
                                                    unsigned lds_byte_off,
                                                    unsigned nelem) {
  unsigned long long ga = (unsigned long long)(uintptr_t)gptr;
  v4u g0;
  g0[0] = 1u;                                   // count=1 (valid), user mode
  g0[1] = lds_byte_off;                         // lds_addr (bytes)
  g0[2] = (unsigned)(ga & 0xffffffffull);       // global_addr[31:0]
  g0[3] = (unsigned)((ga >> 32) & 0x01ffffffull) | (2u << 30);  // [56:32]|type=2
  v8i g1;
  g1[0] = (int)(2u << 16);                      // wg_mask=0, data_size=2 (4B)
  g1[1] = (int)((nelem & 0xffffu) << 16);       // tensor_dim0[15:0]
  g1[2] = (int)(((nelem >> 16) & 0xffffu) | (1u << 16));  // dim0[31:16], dim1=1
  g1[3] = (int)((nelem & 0xffffu) << 16);       // dim1[31:16]=0 | tile_dim0
  g1[4] = 0;                                    // tile_dim1=0, tile_dim2=0
  g1[5] = (int)nelem;                           // tensor_dim0_stride[31:0]
  g1[6] = 0;                                    // stride0[47:32], stride1[15:0]
  g1[7] = 0;                                    // stride1[47:16]
  v4i z4 = {0, 0, 0, 0};                        // groups 2/3: unused (1-D tile)
#if __clang_major__ >= 23
  v8i z8 = {};
  __builtin_amdgcn_tensor_load_to_lds(g0, g1, z4, z4, z8, 0);
#else
  __builtin_amdgcn_tensor_load_to_lds(g0, g1, z4, z4, 0);
#endif
}

// Low 32 bits of a flat LDS pointer are the LDS byte offset (ISA 10.2).
__device__ __forceinline__ unsigned lds_offset_of(const void* p) {
  return (unsigned)(uintptr_t)p;
}

// ---------------------------------------------------------------------------
// Kernel 1: lhs(b,n,t) = sum_f (sum_t' x*W1) * W2[f,t];  rhs(b,t,n) = sum_f x*W3
// One wave32 per (b,n); lanes cover F=64 as 2 floats each (coalesced float2).
// ---------------------------------------------------------------------------
__global__ __launch_bounds__(256) void prep_kernel(
    const float* __restrict__ x, const float* __restrict__ W1,
    const float* __restrict__ W2, const float* __restrict__ W3,
    float* __restrict__ lhs, float* __restrict__ rhs) {
  int gwave = (blockIdx.x * blockDim.x + threadIdx.x) >> 5;
  int lane = threadIdx.x & 31;
  int b = gwave / NN;
  int n = gwave % NN;
  int f0 = lane * 2;

  float r1a = 0.f, r1b = 0.f;
  float w3a = W3[f0], w3b = W3[f0 + 1];
#pragma unroll
  for (int t = 0; t < TT; ++t) {
    const float* p = x + (((size_t)b * TT + t) * NN + n) * FF + f0;
    float2 v = *(const float2*)p;
    float w1 = W1[t];
    r1a += v.x * w1;
    r1b += v.y * w1;
    float part = v.x * w3a + v.y * w3b;
#pragma unroll
    for (int off = 16; off >= 1; off >>= 1) part += __shfl_xor(part, off, 32);
    if (lane == 0) rhs[((size_t)b * TT + t) * NN + n] = part;
  }
#pragma unroll
  for (int t = 0; t < TT; ++t) {
    float part = r1a * W2[f0 * TT + t] + r1b * W2[(f0 + 1) * TT + t];
#pragma unroll
    for (int off = 16; off >= 1; off >>= 1) part += __shfl_xor(part, off, 32);
    if (lane == 0) lhs[((size_t)b * NN + n) * TT + t] = part;
  }
}

// ---------------------------------------------------------------------------
// Kernel 2: S(b,n,m) = sigmoid( sum_t lhs[b,n,t]*rhs[b,t,m] + be[n,m] )
// K=12 -> plain VALU FMAs; one block per (b,n) row, threads stride m.
// ---------------------------------------------------------------------------
__global__ __launch_bounds__(256) void prod_sigmoid_kernel(
    const float* __restrict__ lhs, const float* __restrict__ rhs,
    const float* __restrict__ be, float* __restrict__ S) {
  int bn = blockIdx.x;
  int b = bn / NN;
  int n = bn % NN;
  __shared__ float l[TT];
  if (threadIdx.x < TT) l[threadIdx.x] = lhs[((size_t)b * NN + n) * TT + threadIdx.x];
  __syncthreads();
  const float* rb = rhs + (size_t)b * TT * NN;
  for (int m = threadIdx.x; m < NN; m += blockDim.x) {
    float acc = 0.f;
#pragma unroll
    for (int t = 0; t < TT; ++t) acc += l[t] * rb[t * NN + m];
    float z = acc + be[n * NN + m];
    S[((size_t)b * NN + n) * NN + m] = 1.0f / (1.0f + __expf(-z));
  }
}

// ---------------------------------------------------------------------------
// Kernel 3: E = Ve * S (per batch, 512x512x512 GEMM via v_wmma_f32_16x16x4_f32)
//           then masked softmax rows -> attn.
// Block = 512 threads (16 waves) per (b, 16-row j-tile). Ve strip staged into
// LDS by the TDM; each wave owns a 32-wide m strip (two 16x16 accum tiles).
// WMMA f32 16x16x4 fragment layout (ISA 7.12.2), lane=(l16,hi):
//   A: a[v] = A[l16][2*hi+v]   B: b[v] = B[2*hi+v][l16]   C/D: c[v] = C[8*hi+v][l16]
// ---------------------------------------------------------------------------
__global__ __launch_bounds__(512) void attn_kernel(
    const float* __restrict__ Ve, const float* __restrict__ S,
    const float* __restrict__ A, float* __restrict__ attn) {
  __shared__ float smem[16 * NN];  // 32 KB: Ve strip, then reused for E strip
  int b = blockIdx.x >> 5;
  int jt = blockIdx.x & 31;
  int tid = threadIdx.x;
  int waveId = tid >> 5;
  int lane = tid & 31;
  int l16 = lane & 15;
  int hi = lane >> 4;

  // TDM: DMA Ve rows [jt*16, jt*16+16) (32 KB, contiguous) into LDS.
  if (tid < 32) {
    tdm_load_f32_to_lds(Ve + (size_t)jt * 16 * NN, lds_offset_of(smem), 16 * NN);
    __builtin_amdgcn_s_wait_tensorcnt(0);
  }
  __syncthreads();

  const float* Sb = S + (size_t)b * NN * NN;
  int m0 = waveId * 32;  // this wave's 32-wide m strip
  v8f acc0 = {};
  v8f acc1 = {};
  for (int k0 = 0; k0 < NN; k0 += 4) {
    v2f a;
    a.x = smem[l16 * NN + k0 + 2 * hi + 0];
    a.y = smem[l16 * NN + k0 + 2 * hi + 1];
    const float* bp = Sb + (size_t)(k0 + 2 * hi) * NN;
    __builtin_prefetch(bp + 8 * NN + m0 + l16, 0, 1);
    v2f b0, b1;
    b0.x = bp[m0 + l16];
    b0.y = bp[NN + m0 + l16];
    b1.x = bp[m0 + 16 + l16];
    b1.y = bp[NN + m0 + 16 + l16];
    acc0 = __builtin_amdgcn_wmma_f32_16x16x4_f32(false, a, false, b0, (short)0,
                                                 acc0, false, false);
    acc1 = __builtin_amdgcn_wmma_f32_16x16x4_f32(false, a, false, b1, (short)0,
                                                 acc1, false, false);
  }
  __syncthreads();  // done reading Ve strip; reuse LDS for masked E

  // Apply adjacency mask and park the 16x512 E strip in LDS.
#pragma unroll
  for (int v = 0; v < 8; ++v) {
    int j = v + 8 * hi;
    int jg = jt * 16 + j;
    float e0 = acc0[v] + (-1e16f) * (1.0f - A[jg * NN + m0 + l16]);
    float e1 = acc1[v] + (-1e16f) * (1.0f - A[jg * NN + m0 + 16 + l16]);
    smem[j * NN + m0 + l16] = e0;
    smem[j * NN + m0 + 16 + l16] = e1;
  }
  __syncthreads();

  // Row softmax: wave w owns row j=w (512 elems, 16 per lane).
  {
    int j = waveId;
    const float* row = smem + j * NN;
    float vals[16];
    float mx = -3.0e38f;
#pragma unroll
    for (int i = 0; i < 16; ++i) {
      vals[i] = row[lane + 32 * i];
      mx = fmaxf(mx, vals[i]);
    }
#pragma unroll
    for (int off = 16; off >= 1; off >>= 1) mx = fmaxf(mx, __shfl_xor(mx, off, 32));
    float sum = 0.f;
#pragma unroll
    for (int i = 0; i < 16; ++i) {
      vals[i] = __expf(vals[i] - mx);
      sum += vals[i];
    }
#pragma unroll
    for (int off = 16; off >= 1; off >>= 1) sum += __shfl_xor(sum, off, 32);
    float inv = 1.0f / sum;
    float* orow = attn + ((size_t)b * NN + jt * 16 + j) * NN;
#pragma unroll
    for (int i = 0; i < 16; ++i) orow[lane + 32 * i] = vals[i] * inv;
  }
}

// ---------------------------------------------------------------------------
// Kernel 4: conv = attn * X per (b,t)  (16x64 tile, K=512, WMMA)
//           out  = conv * kernel + bias (16x64, K=64, WMMA), fused via LDS.
// Block = 128 threads (4 waves) per (b, j-tile); attn strip DMA'd to LDS by
// the TDM and reused across all 12 time steps.
// ---------------------------------------------------------------------------
__global__ __launch_bounds__(128) void conv_out_kernel(
    const float* __restrict__ attn, const float* __restrict__ x,
    const float* __restrict__ kernelW, const float* __restrict__ bias,
    float* __restrict__ out) {
  __shared__ float astrip[16 * NN];  // 32 KB attn rows
  __shared__ float c1[16 * FF];      // 4 KB conv tile
  int b = blockIdx.x >> 5;
  int jt = blockIdx.x & 31;
  int tid = threadIdx.x;
  int waveId = tid >> 5;  // 0..3
  int lane = tid & 31;
  int l16 = lane & 15;
  int hi = lane >> 4;

  // TDM: DMA the 16x512 attn strip (32 KB, contiguous) into LDS.
  if (tid < 32) {
    tdm_load_f32_to_lds(attn + ((size_t)b * NN + jt * 16) * NN,
                        lds_offset_of(astrip), 16 * NN);
    __builtin_amdgcn_s_wait_tensorcnt(0);
  }
  __syncthreads();

  int fbase = waveId * 16;  // wave's F tile for conv
  int ubase = waveId * 16;  // wave's U tile for output
  float bv = bias[ubase + l16];

  for (int t = 0; t < TT; ++t) {
    const float* xbt = x + ((size_t)b * TT + t) * NN * FF;
    // conv tile: (16j x 16f), K over 512 nodes
    v8f acc = {};
    for (int k0 = 0; k0 < NN; k0 += 4) {
      v2f a;
      a.x = astrip[l16 * NN + k0 + 2 * hi + 0];
      a.y = astrip[l16 * NN + k0 + 2 * hi + 1];
      const float* bp = xbt + (size_t)(k0 + 2 * hi) * FF + fbase + l16;
      __builtin_prefetch(bp + 8 * FF, 0, 1);
      v2f bb;
      bb.x = bp[0];
      bb.y = bp[FF];
      acc = __builtin_amdgcn_wmma_f32_16x16x4_f32(false, a, false, bb, (short)0,
                                                  acc, false, false);
    }
    __syncthreads();  // previous iteration's c1 consumers are done
#pragma unroll
    for (int v = 0; v < 8; ++v) c1[(v + 8 * hi) * FF + fbase + l16] = acc[v];
    __syncthreads();

    // out tile: (16j x 16u), K over 64 features
    v8f acc2 = {};
#pragma unroll
    for (int k0 = 0; k0 < FF; k0 += 4) {
      v2f a;
      a.x = c1[l16 * FF + k0 + 2 * hi + 0];
      a.y = c1[l16 * FF + k0 + 2 * hi + 1];
      const float* bp = kernelW + (size_t)(k0 + 2 * hi) * UU + ubase + l16;
      v2f bb;
      bb.x = bp[0];
      bb.y = bp[UU];
      acc2 = __builtin_amdgcn_wmma_f32_16x16x4_f32(false, a, false, bb, (short)0,
                                                   acc2, false, false);
    }
    float* op = out + (((size_t)b * TT + t) * NN + jt * 16) * UU;
#pragma unroll
    for (int v = 0; v < 8; ++v) op[(v + 8 * hi) * UU + ubase + l16] = acc2[v] + bv;
  }
}

// ---------------------------------------------------------------------------
extern "C" void kernel_launch(void* const* d_in, const int* in_sizes, int n_in,
                              void* d_out, int out_size, void* d_ws, size_t ws_size,
                              hipStream_t stream) {
  const float* x = (const float*)d_in[0];     // (B,T,N,F)
  const float* A = (const float*)d_in[1];     // (N,N)
  const float* W1 = (const float*)d_in[2];    // (T,1)
  const float* W2 = (const float*)d_in[3];    // (F,T)
  const float* W3 = (const float*)d_in[4];    // (F,1)
  const float* Ve = (const float*)d_in[5];    // (N,N)
  const float* be = (const float*)d_in[6];    // (N,N)
  const float* kw = (const float*)d_in[7];    // (F,U)
  const float* bias = (const float*)d_in[8];  // (U,)
  float* out = (float*)d_out;                 // (B,T,N,U)

  // Workspace layout (floats): lhs | rhs | S | attn  (~137.4 MB total)
  float* ws = (float*)d_ws;
  float* lhs = ws;                                   // B*N*T
  float* rhs = lhs + (size_t)BB * NN * TT;           // B*T*N
  float* S = rhs + (size_t)BB * TT * NN;             // B*N*N
  float* attn = S + (size_t)BB * NN * NN;            // B*N*N

  // 1) lhs / rhs: one wave per (b,n) -> 32768 waves / 8 per block
  prep_kernel<<<(BB * NN) / 8, 256, 0, stream>>>(x, W1, W2, W3, lhs, rhs);
  // 2) product + sigmoid: one block per (b,n)
  prod_sigmoid_kernel<<<BB * NN, 256, 0, stream>>>(lhs, rhs, be, S);
  // 3) Ve*S + mask + softmax: one block per (b, j-tile)
  attn_kernel<<<BB * (NN / 16), 512, 0, stream>>>(Ve, S, A, attn);
  // 4) attn*X then *kernel + bias: one block per (b, j-tile)
  conv_out_kernel<<<BB * (NN / 16), 128, 0, stream>>>(attn, x, kw, bias, out);
}